// LJ_33251636806122
// MI455X (gfx1250) — compile-verified
//
#include <hip/hip_runtime.h>

// MI455X (gfx1250) Lennard-Jones energy/force/virial.
// Memory-bound: ~25 MB traffic -> ~1.1 us at 23.3 TB/s. One thread per atom,
// 6 outgoing edges each (edge list layout: edge k*N+a has i==a). Reductions
// are two-phase (no float atomics -> deterministic); intra-block column sums
// use V_WMMA_F32_16X16X4_F32 with a ones A-matrix as a reduction engine.

typedef __attribute__((ext_vector_type(2))) float v2f;
typedef __attribute__((ext_vector_type(8))) float v8f;

#define TPB 256

// Wave-0-only: column sums of a 128x16 float table in LDS via chained WMMA
// D += ones(16x4) x B(4x16). With A==1, D[m][n] = sum_k B[k][n] (+C), so every
// D row holds the running column totals; result is invariant to which K each
// B row maps to. Returns colsum for column (lane & 15).
__device__ __forceinline__ float wave0_wmma_colsum_128(const float* smem) {
    const int lane = threadIdx.x & 31;
    const int col  = lane & 15;
    const int half = lane >> 4;                 // lanes 16-31 feed K=2,3
    v2f a; a.x = 1.0f; a.y = 1.0f;              // A = ones(16x4)
    v8f d = {0.f, 0.f, 0.f, 0.f, 0.f, 0.f, 0.f, 0.f};
    #pragma unroll 4
    for (int r0 = 0; r0 < 128; r0 += 4) {
        v2f b;
        b.x = smem[(r0 + 2 * half)     * 16 + col];   // B rows K = 2*half
        b.y = smem[(r0 + 2 * half + 1) * 16 + col];   //          K = 2*half+1
        d = __builtin_amdgcn_wmma_f32_16x16x4_f32(
                /*neg_a=*/false, a, /*neg_b=*/false, b,
                /*c_mod=*/(short)0, d, /*reuse_a=*/false, /*reuse_b=*/false);
    }
    return d[0];    // row M=0 (and every row) = column totals
}

__global__ void lj_forces_kernel(const float* __restrict__ pos,
                                 const float* __restrict__ cell,
                                 const float* __restrict__ cutp,
                                 const int*   __restrict__ edst,   // dst row of edge_index
                                 float* __restrict__ force_out,    // d_out + 1
                                 float* __restrict__ ws,           // [numBlocks][8]
                                 int N) {
    __shared__ float smem[TPB * 8];
    const int tid = threadIdx.x;
    const int a   = blockIdx.x * TPB + tid;

    const float cut = cutp[0];
    const float dx = cell[0], dy = cell[1], dz = cell[2];
    const float c2  = cut * cut;
    const float icc = 1.0f / c2;
    const float ic3 = icc * icc * icc;
    const float c0  = -4.0f * (ic3 * ic3 - ic3);   // shift so E(cutoff)=0

    float part[8] = {0.f, 0.f, 0.f, 0.f, 0.f, 0.f, 0.f, 0.f};

    if (a < N) {
        const float pax = pos[3 * a + 0];
        const float pay = pos[3 * a + 1];
        const float paz = pos[3 * a + 2];
        float fx = 0.f, fy = 0.f, fz = 0.f;
        float e = 0.f;
        float wxx = 0.f, wyy = 0.f, wzz = 0.f, wxy = 0.f, wxz = 0.f, wyz = 0.f;

        #pragma unroll
        for (int k = 0; k < 6; ++k) {
            const int j = edst[(size_t)k * (size_t)N + (size_t)a];
            float vx = pos[3 * j + 0] - pax;
            float vy = pos[3 * j + 1] - pay;
            float vz = pos[3 * j + 2] - paz;
            // minimum-image offsets (exactly as reference: masks vs +/-cutoff)
            vx += (vx > cut ? -dx : 0.f) + (vx < -cut ? dx : 0.f);
            vy += (vy > cut ? -dy : 0.f) + (vy < -cut ? dy : 0.f);
            vz += (vz > cut ? -dz : 0.f) + (vz < -cut ? dz : 0.f);

            const float r2   = vx * vx + vy * vy + vz * vz;
            const float inv  = 1.0f / r2;
            const float inv3 = inv * inv * inv;
            const float inv6 = inv3 * inv3;

            e += 4.0f * (inv6 - inv3) + c0;

            const float h  = (48.0f * inv6 - 24.0f * inv3) * inv; // 48 r2^-7 - 24 r2^-4
            const float fs = -2.0f * h;   // force = 2*sum g*v, g = -h; reverse edge exact
            fx += fs * vx; fy += fs * vy; fz += fs * vz;
            // virial = -dE/dstrain = sum h * (v (x) v)
            wxx += h * vx * vx; wyy += h * vy * vy; wzz += h * vz * vz;
            wxy += h * vx * vy; wxz += h * vx * vz; wyz += h * vy * vz;
        }

        force_out[3 * a + 0] = fx;
        force_out[3 * a + 1] = fy;
        force_out[3 * a + 2] = fz;

        part[0] = e;
        part[1] = wxx; part[2] = wyy; part[3] = wzz;
        part[4] = wxy; part[5] = wxz; part[6] = wyz;
    }

    #pragma unroll
    for (int c = 0; c < 8; ++c) smem[tid * 8 + c] = part[c];
    __syncthreads();

    if (tid < 32) {   // full wave32, EXEC all-ones inside -> WMMA legal
        float s = wave0_wmma_colsum_128(smem);
        s += __shfl_xor(s, 8, 32);        // merge even/odd-thread column halves
        if (tid < 8)                      // lanes 0..7 hold totals for comps 0..7
            ws[(size_t)blockIdx.x * 8 + (size_t)tid] = s;
    }
}

__global__ void lj_finalize_kernel(const float* __restrict__ ws, int nb,
                                   float* __restrict__ e_out,     // d_out[0]
                                   float* __restrict__ vir_out) { // d_out + 1 + 3N
    __shared__ float smem[TPB * 8];
    const int tid = threadIdx.x;

    float acc[8] = {0.f, 0.f, 0.f, 0.f, 0.f, 0.f, 0.f, 0.f};
    for (int r = tid; r < nb; r += TPB) {
        #pragma unroll
        for (int c = 0; c < 8; ++c) acc[c] += ws[(size_t)r * 8 + c];
    }
    #pragma unroll
    for (int c = 0; c < 8; ++c) smem[tid * 8 + c] = acc[c];
    __syncthreads();

    if (tid < 32) {
        float s = wave0_wmma_colsum_128(smem);
        s += __shfl_xor(s, 8, 32);
        const float e  = __shfl(s, 0, 32);
        const float xx = __shfl(s, 1, 32);
        const float yy = __shfl(s, 2, 32);
        const float zz = __shfl(s, 3, 32);
        const float xy = __shfl(s, 4, 32);
        const float xz = __shfl(s, 5, 32);
        const float yz = __shfl(s, 6, 32);
        if (tid == 0) {
            e_out[0] = e;
            vir_out[0] = xx; vir_out[1] = xy; vir_out[2] = xz;
            vir_out[3] = xy; vir_out[4] = yy; vir_out[5] = yz;
            vir_out[6] = xz; vir_out[7] = yz; vir_out[8] = zz;
        }
    }
}

extern "C" void kernel_launch(void* const* d_in, const int* in_sizes, int n_in,
                              void* d_out, int out_size, void* d_ws, size_t ws_size,
                              hipStream_t stream) {
    const float* pos  = (const float*)d_in[0];   // [N,3]
    const float* cell = (const float*)d_in[1];   // [3]
    const float* cut  = (const float*)d_in[2];   // [1]
    const int*   edge = (const int*)d_in[3];     // [2, 6N]

    const int N = in_sizes[0] / 3;
    const int E = in_sizes[3] / 2;               // 6N edges
    (void)n_in; (void)out_size; (void)ws_size;

    float* out = (float*)d_out;                  // [E | force(3N) | virial(9)]
    float* ws  = (float*)d_ws;                   // [numBlocks][8] block partials

    const int nb = (N + TPB - 1) / TPB;

    lj_forces_kernel<<<nb, TPB, 0, stream>>>(pos, cell, cut,
                                             edge + (size_t)E,  // dst row
                                             out + 1, ws, N);
    lj_finalize_kernel<<<1, TPB, 0, stream>>>(ws, nb,
                                              out,
                                              out + 1 + (size_t)3 * (size_t)N);
}